// nnconv_10703058502291
// MI455X (gfx1250) — compile-verified
//
#include <hip/hip_runtime.h>
#include <hip/hip_bf16.h>

// Problem constants (match reference)
constexpr int NN = 5000;   // nodes
constexpr int EE = 10000;  // edges
constexpr int GG = 200;    // graphs
constexpr float EPSV = 1e-5f;

typedef __attribute__((ext_vector_type(2))) float v2f;
typedef __attribute__((ext_vector_type(8))) float v8f;

static inline int ceildiv(int a, int b) { return (a + b - 1) / b; }

__device__ __forceinline__ void atomic_add_f(float* p, float v) {
    unsafeAtomicAdd(p, v);   // -> global_atomic_add_f32
}

// ---------------------------------------------------------------------------
// Utility kernels
// ---------------------------------------------------------------------------
__global__ void k_zero(float* p, int n) {
    int t = blockIdx.x * blockDim.x + threadIdx.x;
    if (t < n) p[t] = 0.0f;
}

// H[e, k] = relu(edge_attr[e,:] @ w1[:,k] + b1[k]) for k<32 ; H[e,32] = 1 (bias row)
__global__ void k_edge_hidden(const float* __restrict__ ea,
                              const float* __restrict__ w1,
                              const float* __restrict__ b1,
                              float* __restrict__ H) {
    int t = blockIdx.x * blockDim.x + threadIdx.x;
    if (t >= EE * 33) return;
    int e = t / 33, j = t % 33;
    float v;
    if (j == 32) {
        v = 1.0f;
    } else {
        v = b1[j];
        #pragma unroll
        for (int c = 0; c < 3; ++c) v += ea[e * 3 + c] * w1[c * 32 + j];
        v = fmaxf(v, 0.0f);
    }
    H[t] = v;
}

// Pair-interleaved, din-padded weight reorder.
// Logical row kg = k*dinp + i  (k in [0,33), i in [0,dinp)):
//   value = (i<din) ? (k<32 ? w2[k, i*dout+o] : b2[i*dout+o]) : 0
// Stored as Wp[(kg>>1)*(2*dout) + o*2 + (kg&1)]  -> one b64 per lane in the GEMM.
__global__ void k_reorder_w2p(const float* __restrict__ w2,
                              const float* __restrict__ b2,
                              float* __restrict__ Wp,
                              int din, int dinp, int dout) {
    int t = blockIdx.x * blockDim.x + threadIdx.x;
    int krows = 33 * dinp;
    if (t >= krows * dout) return;
    int kg = t / dout, o = t % dout;
    int k = kg / dinp, i = kg % dinp;
    float v = 0.0f;
    if (i < din) {
        if (k < 32) v = w2[k * (din * dout) + i * dout + o];
        else        v = b2[i * dout + o];
    }
    Wp[(kg >> 1) * (2 * dout) + o * 2 + (kg & 1)] = v;
}

__global__ void k_gather(const float* __restrict__ Xin, const int* __restrict__ src,
                         float* __restrict__ Xs, int din) {
    int t = blockIdx.x * blockDim.x + threadIdx.x;
    if (t >= EE * din) return;
    int e = t / din, i = t % din;
    Xs[t] = Xin[src[e] * din + i];
}

__global__ void k_count_edges(const int* __restrict__ dst, float* __restrict__ cntE) {
    int e = blockIdx.x * blockDim.x + threadIdx.x;
    if (e < EE) atomic_add_f(&cntE[dst[e]], 1.0f);
}

__global__ void k_scatter(const float* __restrict__ msg, const int* __restrict__ dst,
                          float* __restrict__ S, int dout) {
    int t = blockIdx.x * blockDim.x + threadIdx.x;
    if (t >= EE * dout) return;
    int e = t / dout, o = t % dout;
    atomic_add_f(&S[dst[e] * dout + o], msg[t]);
}

// Xout = relu(bn(S/cnt + R + bias))
__global__ void k_epilogue(const float* __restrict__ S, const float* __restrict__ R,
                           const float* __restrict__ cntE, const float* __restrict__ bias,
                           const float* __restrict__ g, const float* __restrict__ b,
                           const float* __restrict__ m, const float* __restrict__ v,
                           float* __restrict__ Xout, int dout) {
    int t = blockIdx.x * blockDim.x + threadIdx.x;
    if (t >= NN * dout) return;
    int node = t / dout, o = t % dout;
    float c = fmaxf(cntE[node], 1.0f);
    float val = S[t] / c + R[t] + bias[o];
    val = (val - m[o]) * (g[o] * rsqrtf(v[o] + EPSV)) + b[o];
    Xout[t] = fmaxf(val, 0.0f);
}

// ---------------------------------------------------------------------------
// Fused message GEMM:  msg[E, DOUT] = P @ W,  P[e, k*DINP+i] = H[e,k] * Xs[e,i]
// (i >= DIN rows are zero). A fragments built from registers; B from the
// pair-interleaved Wp buffer (one b64 per lane per WMMA step).
// E % 16 == 0, DOUT % 16 == 0 -> no M/N guards.
// ---------------------------------------------------------------------------
template <int DIN, int DOUT>
__global__ __launch_bounds__(256) void k_msg_gemm(const float* __restrict__ H,   // [E,33]
                                                  const float* __restrict__ Xs,  // [E,DIN]
                                                  const float* __restrict__ Wp,  // paired
                                                  float* __restrict__ msg) {     // [E,DOUT]
    constexpr int DINP = (DIN + 3) & ~3;        // 16 / 64 / 128
    constexpr int NT   = DOUT / 16;             // 4 / 8 / 16
    constexpr int WACT = (NT < 8) ? NT : 8;     // active waves
    constexpr int NPW  = (NT + 7) / 8;          // N tiles per wave (1 or 2)
    constexpr int XSTR = DINP + 1;              // odd stride -> conflict-free

    __shared__ float Hs[16 * 33];               // 33 odd
    __shared__ float Xss[16 * XSTR];

    const int mtile = blockIdx.x;
    const int lane = threadIdx.x & 31;
    const int wave = threadIdx.x >> 5;

    for (int t = threadIdx.x; t < 16 * 33; t += 256)
        Hs[t] = H[mtile * 16 * 33 + t];
    for (int t = threadIdx.x; t < 16 * DINP; t += 256) {
        int m = t / DINP, i = t % DINP;
        Xss[m * XSTR + i] = (i < DIN) ? Xs[(mtile * 16 + m) * DIN + i] : 0.0f;
    }
    __syncthreads();
    if (wave >= WACT) return;

    const int mn  = lane & 15;        // M for A, N for B/C
    const int kh2 = (lane >> 4);      // half-wave: K pair 0 or 1 within each K4
    const int mrow = kh2 * 8;         // C/D: vgpr r -> M = r (+8 for hi lanes)

    const float* hrow = Hs + mn * 33;
    const float* xrow = Xss + mn * XSTR + kh2 * 2;

    // Hoist this lane's Xs values: i = 4*s + kh2*2 + {0,1}
    float xv[DINP / 2];
    #pragma unroll
    for (int s = 0; s < DINP / 4; ++s) {
        xv[2 * s]     = xrow[4 * s];
        xv[2 * s + 1] = xrow[4 * s + 1];
    }

    v8f acc[NPW];
    #pragma unroll
    for (int j = 0; j < NPW; ++j) acc[j] = (v8f){};

    #pragma unroll 1
    for (int k = 0; k < 33; ++k) {
        const float hk = hrow[k];
        // pair row base for this k (even, since DINP is even)
        const float* wk = Wp + (size_t)((k * DINP) >> 1) * (2 * DOUT) + mn * 2;
        #pragma unroll
        for (int s = 0; s < DINP / 4; ++s) {
            v2f a;
            a[0] = hk * xv[2 * s];
            a[1] = hk * xv[2 * s + 1];
            const float* wpp = wk + (size_t)(2 * s + kh2) * (2 * DOUT);
            #pragma unroll
            for (int j = 0; j < NPW; ++j) {
                int nt = wave + j * 8;
                v2f b = *(const v2f*)(wpp + nt * 32);
                acc[j] = __builtin_amdgcn_wmma_f32_16x16x4_f32(false, a, false, b,
                                                               (short)0, acc[j],
                                                               false, false);
            }
        }
    }

    #pragma unroll
    for (int j = 0; j < NPW; ++j) {
        int nt = wave + j * 8;
        float* mp = msg + (size_t)(mtile * 16) * DOUT + nt * 16 + mn;
        #pragma unroll
        for (int r = 0; r < 8; ++r)
            mp[(size_t)(mrow + r) * DOUT] = acc[j][r];
    }
}

// ---------------------------------------------------------------------------
// Dense WMMA GEMM: R[N, DOUT] = X[N, DIN] @ W[DIN, DOUT]   (root matmul)
// ---------------------------------------------------------------------------
template <int DIN, int DOUT>
__global__ __launch_bounds__(256) void k_root_gemm(const float* __restrict__ X,
                                                   const float* __restrict__ W,
                                                   float* __restrict__ R, int nrows) {
    constexpr int KPAD = (DIN + 3) & ~3;
    constexpr int NT = DOUT / 16;
    const int mtile = blockIdx.x;
    const int lane = threadIdx.x & 31;
    const int wave = threadIdx.x >> 5;
    const int mn = lane & 15;
    const int khalf = (lane >> 4) * 2;
    const int mrow = (lane >> 4) * 8;
    const int row = mtile * 16 + mn;
    const bool rok = row < nrows;

    for (int nt = wave; nt < NT; nt += 8) {
        v8f c = {};
        #pragma unroll 8
        for (int kk = 0; kk < KPAD; kk += 4) {
            v2f a, b;
            #pragma unroll
            for (int q = 0; q < 2; ++q) {
                int kg = kk + khalf + q;
                a[q] = (rok && kg < DIN) ? X[row * DIN + kg] : 0.0f;
                b[q] = (kg < DIN) ? W[kg * DOUT + nt * 16 + mn] : 0.0f;
            }
            c = __builtin_amdgcn_wmma_f32_16x16x4_f32(false, a, false, b,
                                                      (short)0, c, false, false);
        }
        #pragma unroll
        for (int r = 0; r < 8; ++r) {
            int gr = mtile * 16 + mrow + r;
            if (gr < nrows) R[gr * DOUT + nt * 16 + mn] = c[r];
        }
    }
}

// ---------------------------------------------------------------------------
// Graph-level kernels
// ---------------------------------------------------------------------------
__global__ void k_count_graph(const int* __restrict__ batch, float* __restrict__ cntG) {
    int v = blockIdx.x * blockDim.x + threadIdx.x;
    if (v < NN) atomic_add_f(&cntG[batch[v]], 1.0f);
}

__global__ void k_graph_accum(const float* __restrict__ h3, const int* __restrict__ batch,
                              float* __restrict__ gsum) {
    int t = blockIdx.x * blockDim.x + threadIdx.x;
    if (t >= NN * 256) return;
    int node = t / 256, o = t % 256;
    atomic_add_f(&gsum[batch[node] * 256 + o], h3[t]);
}

__global__ void k_scan_offsets(const float* __restrict__ cntG, int* __restrict__ offG) {
    if (threadIdx.x == 0 && blockIdx.x == 0) {
        int run = 0;
        for (int gI = 0; gI < GG; ++gI) {
            offG[gI] = run;
            run += (int)(cntG[gI] + 0.5f);
        }
    }
}

__global__ void k_graph_embed(const float* __restrict__ h3, const int* __restrict__ atom_num,
                              const int* __restrict__ offG, const float* __restrict__ gsum,
                              const float* __restrict__ cntG, float* __restrict__ out_ne,
                              float* __restrict__ hnew) {
    int t = blockIdx.x * blockDim.x + threadIdx.x;
    if (t >= GG * 256) return;
    int gI = t / 256, o = t % 256;
    int idx = atom_num[gI] + offG[gI];
    float ne = h3[idx * 256 + o];
    out_ne[t] = ne;
    float ge = gsum[t] / fmaxf(cntG[gI], 1.0f);
    hnew[t] = 0.5f * ge + ne;
}

__global__ void k_final_mlp(const float* __restrict__ hnew, const float* __restrict__ w,
                            const float* __restrict__ b, float* __restrict__ out) {
    int t = blockIdx.x * blockDim.x + threadIdx.x;
    if (t >= GG * 200) return;
    int gI = t / 200, j = t % 200;
    float acc = b[j];
    #pragma unroll 8
    for (int o = 0; o < 256; ++o) acc += hnew[gI * 256 + o] * w[o * 200 + j];
    out[t] = acc;
}

// ---------------------------------------------------------------------------
// Per-layer driver
// ---------------------------------------------------------------------------
template <int DIN, int DOUT>
static void run_layer(const float* Xin, float* Xout, const float* ea,
                      const int* src, const int* dst,
                      const float* w1, const float* b1, const float* w2, const float* b2,
                      const float* root, const float* bias, const float* bng,
                      const float* bnb, const float* bnm, const float* bnv,
                      float* H, float* Wp, float* Xs, float* msg, float* S, float* R,
                      const float* cntE, hipStream_t stream) {
    constexpr int DINP = (DIN + 3) & ~3;
    k_edge_hidden<<<ceildiv(EE * 33, 256), 256, 0, stream>>>(ea, w1, b1, H);
    k_reorder_w2p<<<ceildiv(33 * DINP * DOUT, 256), 256, 0, stream>>>(w2, b2, Wp, DIN, DINP,
                                                                      DOUT);
    k_gather<<<ceildiv(EE * DIN, 256), 256, 0, stream>>>(Xin, src, Xs, DIN);
    k_zero<<<ceildiv(NN * DOUT, 256), 256, 0, stream>>>(S, NN * DOUT);
    k_msg_gemm<DIN, DOUT><<<EE / 16, 256, 0, stream>>>(H, Xs, Wp, msg);
    k_scatter<<<ceildiv(EE * DOUT, 256), 256, 0, stream>>>(msg, dst, S, DOUT);
    k_root_gemm<DIN, DOUT><<<ceildiv(NN, 16), 256, 0, stream>>>(Xin, root, R, NN);
    k_epilogue<<<ceildiv(NN * DOUT, 256), 256, 0, stream>>>(S, R, cntE, bias, bng, bnb, bnm,
                                                            bnv, Xout, DOUT);
}

extern "C" void kernel_launch(void* const* d_in, const int* in_sizes, int n_in,
                              void* d_out, int out_size, void* d_ws, size_t ws_size,
                              hipStream_t stream) {
    (void)in_sizes; (void)n_in; (void)out_size; (void)ws_size;
    const float* x  = (const float*)d_in[0];
    const float* ea = (const float*)d_in[1];
    const int* ei       = (const int*)d_in[2];
    const int* batch    = (const int*)d_in[3];
    const int* atom_num = (const int*)d_in[4];
    const int* src = ei;
    const int* dst = ei + EE;
    auto LP = [&](int l, int j) { return (const float*)d_in[5 + 10 * (l - 1) + j]; };
    const float* mlp_w = (const float*)d_in[35];
    const float* mlp_b = (const float*)d_in[36];

    // workspace carve (~42 MB of fp32)
    float* base = (float*)d_ws;
    float* H    = base; base += EE * 33;
    float* Wp   = base; base += 33 * 128 * 256;   // max paired-weight buffer (layer 3)
    float* Xs   = base; base += EE * 128;
    float* msg  = base; base += EE * 256;
    float* S    = base; base += NN * 256;
    float* R    = base; base += NN * 256;
    float* hA   = base; base += NN * 256;
    float* hB   = base; base += NN * 256;
    float* cntE = base; base += NN;
    float* cntG = base; base += GG;
    float* gsum = base; base += GG * 256;
    float* hnew = base; base += GG * 256;
    int*   offG = (int*)base;

    // edge-destination counts (shared by all layers)
    k_zero<<<ceildiv(NN, 256), 256, 0, stream>>>(cntE, NN);
    k_count_edges<<<ceildiv(EE, 256), 256, 0, stream>>>(dst, cntE);

    run_layer<15, 64>(x, hA, ea, src, dst, LP(1,0), LP(1,1), LP(1,2), LP(1,3), LP(1,4),
                      LP(1,5), LP(1,6), LP(1,7), LP(1,8), LP(1,9),
                      H, Wp, Xs, msg, S, R, cntE, stream);
    run_layer<64, 128>(hA, hB, ea, src, dst, LP(2,0), LP(2,1), LP(2,2), LP(2,3), LP(2,4),
                       LP(2,5), LP(2,6), LP(2,7), LP(2,8), LP(2,9),
                       H, Wp, Xs, msg, S, R, cntE, stream);
    run_layer<128, 256>(hB, hA, ea, src, dst, LP(3,0), LP(3,1), LP(3,2), LP(3,3), LP(3,4),
                        LP(3,5), LP(3,6), LP(3,7), LP(3,8), LP(3,9),
                        H, Wp, Xs, msg, S, R, cntE, stream);
    const float* h3 = hA;

    // graph stage
    float* out0   = (float*)d_out;          // [G, 200]
    float* out_ne = out0 + GG * 200;        // [G, 256]
    k_zero<<<ceildiv(GG, 256), 256, 0, stream>>>(cntG, GG);
    k_zero<<<ceildiv(GG * 256, 256), 256, 0, stream>>>(gsum, GG * 256);
    k_count_graph<<<ceildiv(NN, 256), 256, 0, stream>>>(batch, cntG);
    k_graph_accum<<<ceildiv(NN * 256, 256), 256, 0, stream>>>(h3, batch, gsum);
    k_scan_offsets<<<1, 32, 0, stream>>>(cntG, offG);
    k_graph_embed<<<ceildiv(GG * 256, 256), 256, 0, stream>>>(h3, atom_num, offG, gsum, cntG,
                                                              out_ne, hnew);
    k_final_mlp<<<ceildiv(GG * 200, 256), 256, 0, stream>>>(hnew, mlp_w, mlp_b, out0);
}